// SpikeEncoder_61332132986987
// MI455X (gfx1250) — compile-verified
//
#include <hip/hip_runtime.h>

// Problem constants (match reference setup_inputs)
#define BB 8
#define SS 2048
#define DD 768
#define LN_EPS 1e-5f
#define LEAK 0.95f
#define THRESH 1.0f

typedef __attribute__((ext_vector_type(2))) float v2f;
typedef __attribute__((ext_vector_type(8))) float v8f;

// ---------------------------------------------------------------------------
// Kernel 1: fused embedding gather + positional add + LayerNorm.
// 256 threads = 8 waves; each wave owns 16 rows (16 (b,s) pairs).
// Sweep 1: WMMA f32 16x16x4 with B=ones reduces 16 rows x 4 cols per
//          instruction into row sums (and row sums-of-squares).
// Sweep 2: recompute emb+pos (L2-resident on a 192MB-L2 part), normalize,
//          coalesced stores of x.
// ---------------------------------------------------------------------------
__global__ __launch_bounds__(256) void ln_embed_kernel(
    const int* __restrict__ tok, const float* __restrict__ emb,
    const float* __restrict__ pos, const float* __restrict__ gamma,
    const float* __restrict__ beta, float* __restrict__ xout)
{
    __shared__ float s_mean[8 * 16];
    __shared__ float s_rstd[8 * 16];

    const int lane    = threadIdx.x & 31;
    const int wave    = threadIdx.x >> 5;
    const int rowBase = blockIdx.x * 128 + wave * 16;   // 128 rows per block
    const int rloc    = lane & 15;                      // A-matrix M = lane&15
    const int hk      = (lane >> 4) * 2;                // half-wave K offset

    const int myRow = rowBase + rloc;
    const int myTok = tok[myRow];
    const float* erow = emb + (long long)myTok * DD;
    const float* prow = pos + (long long)(myRow & (SS - 1)) * DD;

    v8f csum = {};
    v8f csq  = {};
    const v2f ones = {1.0f, 1.0f};

    // ---- sweep 1: row-sum / row-sumsq via WMMA (A = data, B = ones) ----
    #pragma unroll 4
    for (int k0 = 0; k0 < DD; k0 += 4) {
        v2f e = *(const v2f*)(erow + k0 + hk);
        v2f p = *(const v2f*)(prow + k0 + hk);
        v2f a  = e + p;
        v2f a2 = a * a;
        // D[m,n] = sum_k A[m,k] * 1  (row sums, replicated over n)
        csum = __builtin_amdgcn_wmma_f32_16x16x4_f32(
                   false, a,  false, ones, (short)0, csum, false, false);
        csq  = __builtin_amdgcn_wmma_f32_16x16x4_f32(
                   false, a2, false, ones, (short)0, csq,  false, false);
    }

    // C/D layout: VGPR r holds M=r (lanes 0-15) / M=8+r (lanes 16-31),
    // replicated across the 16 N-lanes. Lanes 0 and 16 publish stats to LDS.
    if ((lane & 15) == 0) {
        const int half = (lane >> 4) * 8;
        #pragma unroll
        for (int r = 0; r < 8; ++r) {
            float m   = csum[r] * (1.0f / DD);
            float var = csq[r] * (1.0f / DD) - m * m;
            s_mean[wave * 16 + half + r] = m;
            s_rstd[wave * 16 + half + r] = rsqrtf(var + LN_EPS);
        }
    }
    __syncthreads();

    // ---- sweep 2: recompute (L2-hot) + normalize, coalesced stores ----
    for (int r = 0; r < 16; ++r) {
        const int   row = rowBase + r;
        const float m   = s_mean[wave * 16 + r];
        const float rs  = s_rstd[wave * 16 + r];
        const int   t2  = tok[row];                       // uniform -> s_load
        const float* er = emb + (long long)t2 * DD;
        const float* pr = pos + (long long)(row & (SS - 1)) * DD;
        float*       xr = xout + (long long)row * DD;
        #pragma unroll
        for (int c = lane * 2; c < DD; c += 64) {         // 12 iters, 256B/wave
            v2f e  = *(const v2f*)(er + c);
            v2f p  = *(const v2f*)(pr + c);
            v2f g  = *(const v2f*)(gamma + c);
            v2f bt = *(const v2f*)(beta + c);
            v2f val = e + p;
            v2f y   = (val - m) * rs * g + bt;
            *(v2f*)(xr + c) = y;
        }
    }
}

// ---------------------------------------------------------------------------
// Kernel 2: leaky-integrate-and-fire scan over S. One thread per (b,d) chain
// (6144 chains). x reads hit the 192MB L2 (just written by kernel 1).
// Unroll x4 so four loads are outstanding ahead of the serial v chain.
// Spikes stored non-temporally (never re-read).
// ---------------------------------------------------------------------------
__global__ __launch_bounds__(256) void lif_scan_kernel(
    const float* __restrict__ x, float* __restrict__ spikes)
{
    const int t = blockIdx.x * blockDim.x + threadIdx.x;  // 0..6143
    const int b = t / DD;
    const int d = t - b * DD;
    const float* xp = x      + (long long)b * SS * DD + d;
    float*       sp = spikes + (long long)b * SS * DD + d;

    float v = 0.0f;
    for (int s = 0; s < SS; s += 4) {
        float x0 = xp[(long long)(s + 0) * DD];
        float x1 = xp[(long long)(s + 1) * DD];
        float x2 = xp[(long long)(s + 2) * DD];
        float x3 = xp[(long long)(s + 3) * DD];

        float s0, s1, s2, s3;
        v = v * LEAK + x0; s0 = (v >= THRESH) ? 1.0f : 0.0f; v = (v >= THRESH) ? 0.0f : v;
        v = v * LEAK + x1; s1 = (v >= THRESH) ? 1.0f : 0.0f; v = (v >= THRESH) ? 0.0f : v;
        v = v * LEAK + x2; s2 = (v >= THRESH) ? 1.0f : 0.0f; v = (v >= THRESH) ? 0.0f : v;
        v = v * LEAK + x3; s3 = (v >= THRESH) ? 1.0f : 0.0f; v = (v >= THRESH) ? 0.0f : v;

        __builtin_nontemporal_store(s0, &sp[(long long)(s + 0) * DD]);
        __builtin_nontemporal_store(s1, &sp[(long long)(s + 1) * DD]);
        __builtin_nontemporal_store(s2, &sp[(long long)(s + 2) * DD]);
        __builtin_nontemporal_store(s3, &sp[(long long)(s + 3) * DD]);
    }
}

// ---------------------------------------------------------------------------
extern "C" void kernel_launch(void* const* d_in, const int* in_sizes, int n_in,
                              void* d_out, int out_size, void* d_ws, size_t ws_size,
                              hipStream_t stream)
{
    (void)in_sizes; (void)n_in; (void)out_size; (void)d_ws; (void)ws_size;

    const int*   tok   = (const int*)d_in[0];     // [8,2048] int32
    const float* emb   = (const float*)d_in[1];   // [32000,768]
    const float* pos   = (const float*)d_in[2];   // [2048,768]
    const float* gamma = (const float*)d_in[3];   // [768]
    const float* beta  = (const float*)d_in[4];   // [768]

    // d_out = (spikes[B,S,D], x[B,S,D]) concatenated, both f32
    float* spikes = (float*)d_out;
    float* xout   = spikes + (size_t)BB * SS * DD;

    // 16384 rows total, 128 rows per block
    ln_embed_kernel<<<128, 256, 0, stream>>>(tok, emb, pos, gamma, beta, xout);
    // 6144 chains, 256 threads per block
    lif_scan_kernel<<<24, 256, 0, stream>>>(xout, spikes);
}